// TemporalDistortionLoss_31353261261222
// MI455X (gfx1250) — compile-verified
//
#include <hip/hip_runtime.h>

// Problem constants (pred/target: (8, 768, 64) fp32)
#define MD 768            // m == n == 768
#define KD 64
#define NB 8
#define NDIAG (2*MD - 1)  // 1535 anti-diagonals of the cost matrix
#define INFV 1.0e10f
#define GAMMA 0.1f
#define INVG 10.0f

typedef __attribute__((ext_vector_type(2))) float v2f;
typedef __attribute__((ext_vector_type(8))) float v8f;

// softmin3(a,b,c) = -g*logsumexp(-[a,b,c]/g), computed stably around the min.
__device__ __forceinline__ float softmin3f(float a, float b, float c) {
    float mn = fminf(a, fminf(b, c));
    float s = __expf((mn - a) * INVG) + __expf((mn - b) * INVG) + __expf((mn - c) * INVG);
    return mn - GAMMA * __logf(s);
}

// ---------------------------------------------------------------------------
// Kernel 1: row squared-norms for pred & target, and zero the output scalar.
// ---------------------------------------------------------------------------
__global__ void norms_kernel(const float* __restrict__ pred,
                             const float* __restrict__ target,
                             float* __restrict__ np, float* __restrict__ nt,
                             float* __restrict__ out) {
    int idx = blockIdx.x * blockDim.x + threadIdx.x;
    if (idx == 0) out[0] = 0.0f;
    const int total = NB * MD;
    if (idx >= 2 * total) return;
    const float* src = (idx < total) ? (pred + (size_t)idx * KD)
                                     : (target + (size_t)(idx - total) * KD);
    const float4* s4 = (const float4*)src;
    float s = 0.0f;
#pragma unroll
    for (int k = 0; k < KD / 4; ++k) {
        float4 v = s4[k];
        s += v.x * v.x + v.y * v.y + v.z * v.z + v.w * v.w;
    }
    if (idx < total) np[idx] = s;
    else             nt[idx - total] = s;
}

// ---------------------------------------------------------------------------
// Kernel 2: cost matrix via fp32 WMMA (V_WMMA_F32_16X16X4_F32), one wave per
// 16x16 tile; K=64 -> 16 WMMA steps. Output written in DIAGONAL-SKEWED layout:
//   cell (i,j) 1-based  ->  costD[b][(i+j-2)*768 + (i-1)]
// so the DP kernels get fully coalesced loads along anti-diagonals.
// A layout (16x4 f32): lanes 0-15 M=0..15; VGPR0 K=kb, VGPR1 K=kb+1, kb=(lane>=16)?2:0
// B layout (4x16 f32): lanes hold N=lane&15; VGPR0 K=kb, VGPR1 K=kb+1
// C/D (16x16 f32): VGPR r, lanes 0-15 -> M=r, N=lane; lanes 16-31 -> M=r+8.
// ---------------------------------------------------------------------------
__global__ void cost_wmma_kernel(const float* __restrict__ pred,
                                 const float* __restrict__ target,
                                 const float* __restrict__ np,
                                 const float* __restrict__ nt,
                                 float* __restrict__ costD) {
    const int b  = blockIdx.z;
    const int r0 = blockIdx.x * 16;   // pred-row tile base
    const int c0 = blockIdx.y * 16;   // target-row tile base
    const float* P = pred   + (size_t)b * MD * KD;
    const float* T = target + (size_t)b * MD * KD;

    const int lane = threadIdx.x;          // 0..31, full wave, EXEC all ones
    const int ml   = lane & 15;            // row within tile (A) / col (B)
    const int kb   = (lane >> 4) << 1;     // 0 or 2

    v8f acc = {};
#pragma unroll
    for (int kk = 0; kk < KD; kk += 4) {
        v2f a, bb;
        a.x  = P[(size_t)(r0 + ml) * KD + kk + kb];
        a.y  = P[(size_t)(r0 + ml) * KD + kk + kb + 1];
        bb.x = T[(size_t)(c0 + ml) * KD + kk + kb];
        bb.y = T[(size_t)(c0 + ml) * KD + kk + kb + 1];
        acc = __builtin_amdgcn_wmma_f32_16x16x4_f32(
            /*neg_a=*/false, a, /*neg_b=*/false, bb,
            /*c_mod=*/(short)0, acc, /*reuse_a=*/false, /*reuse_b=*/false);
    }

    const float* npb = np + b * MD;
    const float* ntb = nt + b * MD;
    float* CD = costD + (size_t)b * NDIAG * MD;
    const int nn  = lane & 15;
    const int mhi = (lane >> 4) * 8;
#pragma unroll
    for (int r = 0; r < 8; ++r) {
        int gm = r0 + r + mhi;       // 0-based row
        int gn = c0 + nn;            // 0-based col
        float c = fmaxf(npb[gm] + ntb[gn] - 2.0f * acc[r], 0.0f);
        CD[(size_t)(gm + gn) * MD + gm] = c;   // skewed store
    }
}

// ---------------------------------------------------------------------------
// Kernel 3: soft-DTW forward. One workgroup per batch, 768 threads (thread t
// owns row i=t+1). Rolling 3-diagonal LDS buffers; one barrier per diagonal.
// Writes interior R values (the "f" matrix) in the same skewed layout, and
// dist[b] = R[m][n].
// ---------------------------------------------------------------------------
__global__ void __launch_bounds__(MD)
sdtw_forward_kernel(const float* __restrict__ costD,
                    float* __restrict__ fD,
                    float* __restrict__ dist) {
    const int b   = blockIdx.x;
    const int tid = threadIdx.x;   // 0..767
    const int i   = tid + 1;       // 1..768
    const size_t boff = (size_t)b * NDIAG * MD;

    __shared__ float buf[3][MD + 4];   // entries 0..768 used (R rows 0..m)

    for (int x = tid; x < MD + 1; x += MD) {
        buf[0][x] = INFV; buf[1][x] = INFV; buf[2][x] = INFV;
    }
    if (tid == 0) buf[0][0] = 0.0f;    // R[0][0] = 0 on diagonal d=0
    __syncthreads();

    float lastv = INFV;
    for (int d = 2; d <= 2 * MD; ++d) {
        float*       cur = buf[d % 3];         // diagonal d   (overwrites d-3)
        const float* p1  = buf[(d + 2) % 3];   // diagonal d-1
        const float* p2  = buf[(d + 1) % 3];   // diagonal d-2
        const int j = d - i;
        float v = INFV;
        if (j >= 1 && j <= MD) {
            float c  = costD[boff + (size_t)(d - 2) * MD + (i - 1)];
            float sm = softmin3f(p2[i - 1], p1[i - 1], p1[i]);
            v = c + sm;
            fD[boff + (size_t)(d - 2) * MD + (i - 1)] = v;
            if (d < 2 * MD)  // prefetch next diagonal's cost (global_prefetch)
                __builtin_prefetch(&costD[boff + (size_t)(d - 1) * MD + (i - 1)], 0, 3);
        }
        cur[i] = v;
        if (tid == 0) cur[0] = INFV;           // R[0][d] = INF
        if (d == 2 * MD && i == MD) lastv = v; // R[m][n]
        __syncthreads();
    }
    if (i == MD) dist[b] = lastv;
}

// ---------------------------------------------------------------------------
// Kernel 4: soft-DTW backward, fused with the alignment reduction: Rb is never
// materialized; per cell we immediately form a = exp(min(-(f+c+b-dist)/g, 0))
// and accumulate S1 = sum(a), S2 = sum(a * distortion). One workgroup/batch.
// INF-initialized rolling buffers exactly reproduce the reference semantics
// (its Rb[m][n]=0 seed is overwritten on the first diagonal before any read).
// ---------------------------------------------------------------------------
__global__ void __launch_bounds__(MD)
sdtw_backward_kernel(const float* __restrict__ costD,
                     const float* __restrict__ fD,
                     const float* __restrict__ dist,
                     float* __restrict__ out) {
    const int b   = blockIdx.x;
    const int tid = threadIdx.x;
    const int i   = tid + 1;       // 1..768
    const size_t boff = (size_t)b * NDIAG * MD;

    __shared__ float buf[3][MD + 4];   // entries 0..769 (Rb rows 0..m+1)
    __shared__ float red[2][MD];

    for (int x = tid; x < MD + 2; x += MD) {
        buf[0][x] = INFV; buf[1][x] = INFV; buf[2][x] = INFV;
    }
    __syncthreads();

    const float db   = dist[b];
    const float invL = 1.0f / (float)(MD - 1);
    float S1 = 0.0f, S2 = 0.0f;

    for (int d = 2 * MD; d >= 2; --d) {
        float*       cur = buf[d % 3];         // diagonal d   (overwrites d+3)
        const float* q1  = buf[(d + 1) % 3];   // diagonal d+1
        const float* q2  = buf[(d + 2) % 3];   // diagonal d+2
        const int j = d - i;
        float v = INFV;
        if (j >= 1 && j <= MD) {
            const bool iok = (i + 1 <= MD);
            const bool jok = (j + 1 <= MD);
            // skewed cost: cell (a,b) -> (a+b-2)*MD + (a-1)
            float diag  = (iok && jok)
                        ? q2[i + 1] + costD[boff + (size_t)d * MD + i]            : INFV;
            float below = iok
                        ? q1[i + 1] + costD[boff + (size_t)(d - 1) * MD + i]      : INFV;
            float right = jok
                        ? q1[i]     + costD[boff + (size_t)(d - 1) * MD + (i - 1)]: INFV;
            v = softmin3f(diag, below, right);
            // fused alignment accumulation for cell (i,j)
            float cv = costD[boff + (size_t)(d - 2) * MD + (i - 1)];
            float fv = fD   [boff + (size_t)(d - 2) * MD + (i - 1)];
            float lp = fminf(-(fv + cv + v - db) * INVG, 0.0f);
            float a  = __expf(lp);
            float ip = (float)(i - 1) * invL;
            float jp = (float)(j - 1) * invL;
            float dd = ip - jp; dd *= dd;
            S1 += a;
            S2 += a * dd;
            if (d > 2)  // prefetch next (smaller) diagonal's cost
                __builtin_prefetch(&costD[boff + (size_t)(d - 3) * MD + (i - 1)], 0, 3);
        }
        cur[i] = v;
        if (tid == 0) cur[0] = INFV;           // Rb row 0 boundary
        __syncthreads();
    }

    // block reduction of S1, S2 (768 = 3*256)
    red[0][tid] = S1; red[1][tid] = S2;
    __syncthreads();
    if (tid < 256) {
        red[0][tid] += red[0][tid + 256] + red[0][tid + 512];
        red[1][tid] += red[1][tid + 256] + red[1][tid + 512];
    }
    __syncthreads();
    for (int s = 128; s > 0; s >>= 1) {
        if (tid < s) {
            red[0][tid] += red[0][tid + s];
            red[1][tid] += red[1][tid + s];
        }
        __syncthreads();
    }
    if (tid == 0) {
        float t = red[0][0], w = red[1][0];
        // A normalized by total only if total > 1e-8; result = sum(A*distortion)
        float loss = (t > 1e-8f) ? (w / t) * (float)MD : w;
        atomicAdd(out, loss * (1.0f / (float)NB));   // mean over batches
    }
}

// ---------------------------------------------------------------------------
extern "C" void kernel_launch(void* const* d_in, const int* in_sizes, int n_in,
                              void* d_out, int out_size, void* d_ws, size_t ws_size,
                              hipStream_t stream) {
    const float* pred   = (const float*)d_in[0];
    const float* target = (const float*)d_in[1];
    float* out = (float*)d_out;
    float* ws  = (float*)d_ws;

    const size_t CSZ = (size_t)NB * NDIAG * MD;   // skewed matrix size (floats)
    float* costD = ws;                  // 8 * 1535*768 floats  (~37.7 MB)
    float* fD    = ws + CSZ;            // 8 * 1535*768 floats  (~37.7 MB)
    float* np    = ws + 2 * CSZ;        // 8*768
    float* nt    = np + NB * MD;        // 8*768
    float* dist  = nt + NB * MD;        // 8

    norms_kernel<<<(2 * NB * MD + 255) / 256, 256, 0, stream>>>(pred, target, np, nt, out);
    cost_wmma_kernel<<<dim3(MD / 16, MD / 16, NB), 32, 0, stream>>>(pred, target, np, nt, costD);
    sdtw_forward_kernel<<<NB, MD, 0, stream>>>(costD, fD, dist);
    sdtw_backward_kernel<<<NB, MD, 0, stream>>>(costD, fD, dist, out);
}